// GATModel_84473416778472
// MI455X (gfx1250) — compile-verified
//
#include <hip/hip_runtime.h>
#include <math.h>

// ---------------- constants (match reference) ----------------
#define HIDC 768
#define NHC  12
#define DKC  64
#define BC   8
#define NUC  512
#define NQC  64
#define SLOPEC 0.2f
#define NEGV  (-1e30f)

typedef __attribute__((ext_vector_type(2))) float v2f;
typedef __attribute__((ext_vector_type(8))) float v8f;
typedef __attribute__((ext_vector_type(4))) int   v4i;

#define AS1 __attribute__((address_space(1)))
#define AS3 __attribute__((address_space(3)))

#if defined(__gfx1250__) && __has_builtin(__builtin_amdgcn_global_load_async_to_lds_b128)
#define HAVE_ASYNC_LDS 1
#else
#define HAVE_ASYNC_LDS 0
#endif

// Exact-f32 WMMA: D = A(16x4) x B(4x16) + C(16x16), wave32.
__device__ __forceinline__ v8f wmma_f32(v2f a, v2f b, v8f c) {
    return __builtin_amdgcn_wmma_f32_16x16x4_f32(false, a, false, b, (short)0, c,
                                                 false, false);
}

// 16-byte global -> LDS copy; async on CDNA5 (ASYNCcnt), sync fallback.
__device__ __forceinline__ void cp16(float* l, const float* g) {
#if HAVE_ASYNC_LDS
    __builtin_amdgcn_global_load_async_to_lds_b128((AS1 v4i*)g, (AS3 v4i*)l,
                                                   0, 0);
#else
    *(float4*)l = *(const float4*)g;
#endif
}

__device__ __forceinline__ void async_wait() {
#if HAVE_ASYNC_LDS
#if __has_builtin(__builtin_amdgcn_s_wait_asynccnt)
    __builtin_amdgcn_s_wait_asynccnt(0);
#else
    asm volatile("s_wait_asynccnt 0x0" ::: "memory");
#endif
#endif
}

// =====================================================================
// GEMM: Y[M,N] = X[M,K] @ W[N,K]^T + bias (+ optional exact GELU)
// Block = 256 thr = 8 waves; block tile 128(M) x 64(N); K staged through
// double-buffered LDS in KT=32 chunks with async b128 copies.
// Row stride 36 floats: 16B-aligned staging + conflict-free b64 reads.
// M % 128 == 0, N % 64 == 0, K % 32 == 0.
// =====================================================================
#define KT 32
#define LSTR 36

__device__ __forceinline__ void stage_tiles(const float* __restrict__ Xb,
                                            const float* __restrict__ Wb,
                                            int K, int k0,
                                            float* sx, float* sw, int tid)
{
    // X tile: 128 rows x 32 floats = 1024 float4, 4 per thread (coalesced)
    #pragma unroll
    for (int i = 0; i < 4; ++i) {
        const int fidx = (tid + i * 256) * 4;
        const int row = fidx >> 5, col = fidx & 31;
        cp16(sx + row * LSTR + col, Xb + (size_t)row * K + k0 + col);
    }
    // W tile: 64 rows x 32 floats = 512 float4, 2 per thread
    #pragma unroll
    for (int i = 0; i < 2; ++i) {
        const int fidx = (tid + i * 256) * 4;
        const int row = fidx >> 5, col = fidx & 31;
        cp16(sw + row * LSTR + col, Wb + (size_t)row * K + k0 + col);
    }
}

__global__ __launch_bounds__(256) void gemm_wmma_kernel(
    const float* __restrict__ X, const float* __restrict__ W,
    const float* __restrict__ bias, float* __restrict__ Y,
    int M, int N, int K, int act)
{
    __shared__ float sX[2][128 * LSTR];   // 36 KB
    __shared__ float sW[2][64 * LSTR];    // 18 KB
    const int tid  = threadIdx.x;
    const int lane = tid & 31;
    const int wave = tid >> 5;
    const int lr   = lane & 15;
    const int hi   = lane >> 4;           // K-half select (ISA A/B layout)
    const int row0 = blockIdx.x * 128 + wave * 16;
    const int col0 = blockIdx.y * 64;

    const float* Xb = X + (size_t)blockIdx.x * 128 * K;
    const float* Wb = W + (size_t)col0 * K;

    v8f acc[4] = {v8f{}, v8f{}, v8f{}, v8f{}};

    stage_tiles(Xb, Wb, K, 0, sX[0], sW[0], tid);
    async_wait();
    __syncthreads();

    const int nchunks = K / KT;
    int buf = 0;
    for (int c = 0; c < nchunks; ++c) {
        if (c + 1 < nchunks)   // overlaps with compute below (ASYNCcnt)
            stage_tiles(Xb, Wb, K, (c + 1) * KT, sX[buf ^ 1], sW[buf ^ 1], tid);

        const float* xa = &sX[buf][(wave * 16 + lr) * LSTR];
        const float* wa = &sW[buf][lr * LSTR];
        #pragma unroll
        for (int k = 0; k < KT; k += 4) {
            const int ko = k + hi * 2;
            v2f a = *(const v2f*)(xa + ko);
            #pragma unroll
            for (int j = 0; j < 4; ++j) {
                v2f b = *(const v2f*)(wa + j * 16 * LSTR + ko);
                acc[j] = wmma_f32(a, b, acc[j]);
            }
        }
        async_wait();
        __syncthreads();
        buf ^= 1;
    }

    #pragma unroll
    for (int j = 0; j < 4; ++j) {
        const int col = col0 + j * 16 + lr;
        const float bv = bias ? bias[col] : 0.0f;
        #pragma unroll
        for (int r = 0; r < 8; ++r) {
            const int t = row0 + r + hi * 8;          // C/D row mapping
            float v = acc[j][r] + bv;
            if (act == 1)                              // exact GELU
                v = 0.5f * v * (1.0f + erff(v * 0.70710678118654752f));
            Y[(size_t)t * N + col] = v;
        }
    }
}

// =====================================================================
// el/er head dots: out[(b*NH+h)*N + n] = sum_d F[b,n,h*64+d] * vec[h*64+d]
// =====================================================================
__global__ void dot_head_kernel(const float* __restrict__ F,
                                const float* __restrict__ vec,
                                float* __restrict__ out, int N, int total)
{
    int i = blockIdx.x * blockDim.x + threadIdx.x;
    if (i >= total) return;
    const int n  = i % N;
    const int bh = i / N;
    const int h  = bh % NHC;
    const int b  = bh / NHC;
    const float* f = F + ((size_t)b * N + n) * HIDC + h * DKC;
    const float* v = vec + h * DKC;
    float s = 0.0f;
    #pragma unroll 8
    for (int d = 0; d < DKC; ++d) s += f[d] * v[d];
    out[i] = s;
}

// =====================================================================
// masked score: out[b,h,t,s] = adj[b,s,t]>0 ? lrelu(er+el+pri) : NEG
// =====================================================================
__global__ void score_kernel(const float* __restrict__ er,
                             const float* __restrict__ el,
                             const float* __restrict__ pri,
                             const float* __restrict__ adj,
                             float* __restrict__ out,
                             int Nd, int Ns, int total)
{
    int i = blockIdx.x * blockDim.x + threadIdx.x;
    if (i >= total) return;
    const int s    = i % Ns;
    const int rest = i / Ns;
    const int t    = rest % Nd;
    const int bh   = rest / Nd;
    const int h    = bh % NHC;
    const int b    = bh / NHC;
    float sc = er[(size_t)bh * Nd + t] + el[(size_t)bh * Ns + s] + pri[h];
    sc = sc > 0.0f ? sc : SLOPEC * sc;
    const float m = adj[((size_t)b * Ns + s) * Nd + t];
    out[i] = (m > 0.0f) ? sc : NEGV;
}

// =====================================================================
// per-row softmax stats: rmax, 1/sum(exp(x-max)).  One wave per row.
// (All-masked rows: max=NEG, exps=1, sum=Ns -> uniform 1/Ns, matches jnp.)
// =====================================================================
__global__ __launch_bounds__(256) void rowstats_kernel(
    const float* __restrict__ masked, float* __restrict__ rmax,
    float* __restrict__ rinv, int Ns, int rowsTotal)
{
    const int row  = blockIdx.x * 8 + (threadIdx.x >> 5);
    const int lane = threadIdx.x & 31;
    if (row >= rowsTotal) return;
    const float* p = masked + (size_t)row * Ns;
    float m = -INFINITY;
    for (int s = lane; s < Ns; s += 32) m = fmaxf(m, p[s]);
    for (int o = 16; o > 0; o >>= 1) m = fmaxf(m, __shfl_xor(m, o, 32));
    float sum = 0.0f;
    for (int s = lane; s < Ns; s += 32) sum += __expf(p[s] - m);
    for (int o = 16; o > 0; o >>= 1) sum += __shfl_xor(sum, o, 32);
    if (lane == 0) { rmax[row] = m; rinv[row] = 1.0f / sum; }
}

// =====================================================================
// WMMA aggregation with fused softmax (register-double-buffered):
//   UPD[b, t, h*64+d] (+=) w * sum_s softmax(masked)[b,h,t,s] * FS[b,s,h*64+d]
// Block = 128 thr = 4 waves -> 64 rows/block. grid = (Nd/64, B*NH).
// mode: 0 -> w = softmax(cew)[0]; 1 -> softmax(cew)[1]; 2 -> w = 1.
// =====================================================================
__global__ __launch_bounds__(128) void attn_agg_kernel(
    const float* __restrict__ masked, const float* __restrict__ rmax,
    const float* __restrict__ rinv,   const float* __restrict__ FS,
    float* __restrict__ UPD, int Nd, int Ns,
    const float* __restrict__ cew, int mode, int accum)
{
    const int bh   = blockIdx.y;
    const int b    = bh / NHC;
    const int h    = bh % NHC;
    const int lane = threadIdx.x & 31;
    const int wave = threadIdx.x >> 5;
    const int lr   = lane & 15;
    const int hi   = lane >> 4;
    const int t0   = blockIdx.x * 64 + wave * 16;
    if (t0 >= Nd) return;                // wave-uniform

    float w = 1.0f;
    if (mode < 2) {
        const float e0 = __expf(cew[0]);
        const float e1 = __expf(cew[1]);
        w = (mode == 0 ? e0 : e1) / (e0 + e1);
    }

    const int   row  = t0 + lr;
    const float mx   = rmax[(size_t)bh * Nd + row];
    const float iv   = rinv[(size_t)bh * Nd + row];
    const float* mr  = masked + ((size_t)bh * Nd + row) * Ns;
    const float* fpan = FS + (size_t)b * Ns * HIDC + h * DKC;

    v8f acc[4] = {v8f{}, v8f{}, v8f{}, v8f{}};

    // prologue fragment loads
    float2 mv = *(const float2*)(mr + hi * 2);
    float bc[8];
    #pragma unroll
    for (int j = 0; j < 4; ++j) {
        const int col = j * 16 + lr;
        bc[2 * j]     = fpan[(size_t)(hi * 2) * HIDC + col];
        bc[2 * j + 1] = fpan[(size_t)(hi * 2 + 1) * HIDC + col];
    }

    for (int k = 0; k < Ns; k += 4) {
        float2 nmv = make_float2(0.0f, 0.0f);
        float bn[8] = {0, 0, 0, 0, 0, 0, 0, 0};
        const bool more = (k + 4) < Ns;  // wave-uniform
        if (more) {                      // prefetch next iteration's fragments
            const int kn = k + 4 + hi * 2;
            nmv = *(const float2*)(mr + kn);
            __builtin_prefetch(mr + kn + 64, 0, 3);
            #pragma unroll
            for (int j = 0; j < 4; ++j) {
                const int col = j * 16 + lr;
                bn[2 * j]     = fpan[(size_t)kn * HIDC + col];
                bn[2 * j + 1] = fpan[(size_t)(kn + 1) * HIDC + col];
            }
        }
        v2f a;
        a[0] = __expf(mv.x - mx) * iv;
        a[1] = __expf(mv.y - mx) * iv;
        #pragma unroll
        for (int j = 0; j < 4; ++j) {
            v2f bb; bb[0] = bc[2 * j]; bb[1] = bc[2 * j + 1];
            acc[j] = wmma_f32(a, bb, acc[j]);
        }
        mv = nmv;
        #pragma unroll
        for (int j = 0; j < 8; ++j) bc[j] = bn[j];
    }

    #pragma unroll
    for (int j = 0; j < 4; ++j) {
        const int col = j * 16 + lr;
        #pragma unroll
        for (int r = 0; r < 8; ++r) {
            const int t = t0 + r + hi * 8;
            const size_t oi = ((size_t)b * Nd + t) * HIDC + h * DKC + col;
            const float v = w * acc[j][r];
            if (accum) UPD[oi] += v; else UPD[oi] = v;
        }
    }
}

// =====================================================================
// fused skip + LayerNorm: one wave per 768-element row (24 elems/lane)
// =====================================================================
__global__ __launch_bounds__(256) void skip_ln_kernel(
    const float* __restrict__ trans, const float* __restrict__ hold,
    const float* __restrict__ skipP, const float* __restrict__ lng,
    const float* __restrict__ lnb, float* __restrict__ hnew, int rows)
{
    const int row  = blockIdx.x * 8 + (threadIdx.x >> 5);
    const int lane = threadIdx.x & 31;
    if (row >= rows) return;
    const float alpha = 1.0f / (1.0f + __expf(-skipP[0]));
    const float* tr = trans + (size_t)row * HIDC;
    const float* ho = hold + (size_t)row * HIDC;
    float y[24];
    float sum = 0.0f;
    #pragma unroll
    for (int j = 0; j < 24; ++j) {
        const int d = lane + j * 32;
        const float v = tr[d] * alpha + ho[d] * (1.0f - alpha);
        y[j] = v; sum += v;
    }
    for (int o = 16; o > 0; o >>= 1) sum += __shfl_xor(sum, o, 32);
    const float mean = sum * (1.0f / 768.0f);
    float vs = 0.0f;
    #pragma unroll
    for (int j = 0; j < 24; ++j) { const float d = y[j] - mean; vs += d * d; }
    for (int o = 16; o > 0; o >>= 1) vs += __shfl_xor(vs, o, 32);
    const float inv = rsqrtf(vs * (1.0f / 768.0f) + 1e-5f);
    float* outp = hnew + (size_t)row * HIDC;
    #pragma unroll
    for (int j = 0; j < 24; ++j) {
        const int d = lane + j * 32;
        outp[d] = (y[j] - mean) * inv * lng[d] + lnb[d];
    }
}

// =====================================================================
// Host orchestration
// =====================================================================
// Input indices: jax pytree order (dict keys sorted; layer list in order)
enum { I_ADAPT_W = 0, I_ADAPT_B = 1, I_L0 = 2, I_LSTRIDE = 14,
       I_OUT_W = 30, I_OUT_B = 31, I_FU = 32, I_FQ = 33,
       I_AUU = 34, I_AQU = 35, I_AUQ = 36 };
// per-layer leaf offsets (alphabetical within layer dict)
enum { L_AW = 0, L_AB = 1, L_ATTN_L = 2, L_ATTN_R = 3, L_CEWQ = 4, L_CEWU = 5,
       L_DW = 6, L_DB = 7, L_LNB = 8, L_LNG = 9, L_PRI = 10, L_SKIP = 11,
       L_SW = 12, L_SB = 13 };

extern "C" void kernel_launch(void* const* d_in, const int* in_sizes, int n_in,
                              void* d_out, int out_size, void* d_ws, size_t ws_size,
                              hipStream_t stream)
{
    (void)in_sizes; (void)n_in; (void)out_size; (void)ws_size;
    auto in = [&](int i) { return (const float*)d_in[i]; };
    float* out = (float*)d_out;
    float* ws  = (float*)d_ws;

    const size_t HU = (size_t)BC * NUC * HIDC;   // 3,145,728
    const size_t HQ = (size_t)BC * NQC * HIDC;   //   393,216
    const size_t RU = (size_t)BC * NHC * NUC;    //    49,152
    const size_t RQ = (size_t)BC * NHC * NQC;    //     6,144
    const int MU = BC * NUC;                     // 4096 rows
    const int MQ = BC * NQC;                     //  512 rows

    size_t off = 0;
    auto alloc = [&](size_t n) { float* p = ws + off; off += n; return p; };
    float* hu0  = alloc(HU);  float* hu1  = alloc(HU);
    float* hq0  = alloc(HQ);  float* hq1  = alloc(HQ);
    float* fs_u = alloc(HU);  float* fd_u = alloc(HU);
    float* fs_q = alloc(HQ);  float* fd_q = alloc(HQ);
    float* el0 = alloc(RU); float* el1 = alloc(RQ); float* el2 = alloc(RU);
    float* er0 = alloc(RU); float* er1 = alloc(RU); float* er2 = alloc(RQ);
    float* mx0 = alloc(RU); float* iv0 = alloc(RU);
    float* mx1 = alloc(RU); float* iv1 = alloc(RU);
    float* mx2 = alloc(RQ); float* iv2 = alloc(RQ);
    // aliases (buffers free by the time these are written)
    float* upd_u = fd_u;  float* upd_q = fd_q;
    float* trans_u = fs_u; float* trans_q = fs_q;

    // output regions
    float* o_utt = out;                               // [8,512,768]
    float* o_q   = out + HU;                          // [8,64,768]
    float* m_uu  = o_q + HQ;                          // [8,12,512,512]
    float* m_qu  = m_uu + (size_t)BC * NHC * NUC * NUC;   // [8,12,512,64]
    float* m_uq  = m_qu + (size_t)BC * NHC * NUC * NQC;   // [8,12,64,512]

    auto gemm = [&](const float* X, const float* W, const float* Bp, float* Y,
                    int M, int N, int K, int act) {
        dim3 g((unsigned)(M / 128), (unsigned)(N / 64));
        gemm_wmma_kernel<<<g, 256, 0, stream>>>(X, W, Bp, Y, M, N, K, act);
    };
    auto dot = [&](const float* F, const float* vec, float* o, int N) {
        const int total = BC * NHC * N;
        dot_head_kernel<<<total / 256, 256, 0, stream>>>(F, vec, o, N, total);
    };
    auto score = [&](const float* er, const float* el, const float* pri,
                     const float* adj, float* o, int Nd, int Ns) {
        const int total = BC * NHC * Nd * Ns;
        score_kernel<<<total / 256, 256, 0, stream>>>(er, el, pri, adj, o, Nd, Ns, total);
    };
    auto rowstats = [&](const float* msk, float* mx, float* iv, int Nd, int Ns) {
        const int rows = BC * NHC * Nd;
        rowstats_kernel<<<rows / 8, 256, 0, stream>>>(msk, mx, iv, Ns, rows);
    };
    auto agg = [&](const float* msk, const float* mx, const float* iv,
                   const float* FS, float* UPD, int Nd, int Ns,
                   const float* cew, int mode, int accum) {
        dim3 g((unsigned)(Nd / 64), (unsigned)(BC * NHC));
        attn_agg_kernel<<<g, 128, 0, stream>>>(msk, mx, iv, FS, UPD, Nd, Ns,
                                               cew, mode, accum);
    };
    auto skipln = [&](const float* tr, const float* ho, const float* sk,
                      const float* g, const float* b, float* hn, int rows) {
        skip_ln_kernel<<<rows / 8, 256, 0, stream>>>(tr, ho, sk, g, b, hn, rows);
    };

    // ---- adapter: h = gelu(feats @ adapt_W[t]^T + adapt_b[t]) ----
    gemm(in(I_FU), in(I_ADAPT_W),               in(I_ADAPT_B),        hu0, MU, HIDC, HIDC, 1);
    gemm(in(I_FQ), in(I_ADAPT_W) + HIDC * HIDC, in(I_ADAPT_B) + HIDC, hq0, MQ, HIDC, HIDC, 1);

    const float* adj_uu = in(I_AUU);
    const float* adj_qu = in(I_AQU);
    const float* adj_uq = in(I_AUQ);

    float* hu_in = hu0; float* hq_in = hq0;
    float* hu_out = hu1; float* hq_out = hq1;

    for (int l = 0; l < 2; ++l) {
        const int lb = I_L0 + l * I_LSTRIDE;
        const float* aW  = in(lb + L_AW);     const float* aB  = in(lb + L_AB);
        const float* aL  = in(lb + L_ATTN_L); const float* aR  = in(lb + L_ATTN_R);
        const float* cewU = in(lb + L_CEWU);
        const float* dW  = in(lb + L_DW);     const float* dB  = in(lb + L_DB);
        const float* lnB = in(lb + L_LNB);    const float* lnG = in(lb + L_LNG);
        const float* pri = in(lb + L_PRI);    const float* sk  = in(lb + L_SKIP);
        const float* sW  = in(lb + L_SW);     const float* sB  = in(lb + L_SB);

        // projections (fs_u shared by edges uu & uq; fd_u by uu & qu)
        gemm(hu_in, sW,               sB,        fs_u, MU, HIDC, HIDC, 0);
        gemm(hu_in, dW,               dB,        fd_u, MU, HIDC, HIDC, 0);
        gemm(hq_in, sW + HIDC * HIDC, sB + HIDC, fs_q, MQ, HIDC, HIDC, 0);
        gemm(hq_in, dW + HIDC * HIDC, dB + HIDC, fd_q, MQ, HIDC, HIDC, 0);

        // per-head attention dots
        dot(fs_u, aL + 0 * HIDC, el0, NUC);
        dot(fs_q, aL + 1 * HIDC, el1, NQC);
        dot(fs_u, aL + 2 * HIDC, el2, NUC);
        dot(fd_u, aR + 0 * HIDC, er0, NUC);
        dot(fd_u, aR + 1 * HIDC, er1, NUC);
        dot(fd_q, aR + 2 * HIDC, er2, NQC);

        // masked scores (written straight into d_out; last layer's persist)
        score(er0, el0, pri + 0,  adj_uu, m_uu, NUC, NUC);
        score(er1, el1, pri + 12, adj_qu, m_qu, NUC, NQC);
        score(er2, el2, pri + 24, adj_uq, m_uq, NQC, NUC);

        // softmax row stats
        rowstats(m_uu, mx0, iv0, NUC, NUC);
        rowstats(m_qu, mx1, iv1, NUC, NQC);
        rowstats(m_uq, mx2, iv2, NQC, NUC);

        // fused softmax + WMMA aggregation, cross-edge-weighted
        agg(m_uu, mx0, iv0, fs_u, upd_u, NUC, NUC, cewU, 0, 0);
        agg(m_qu, mx1, iv1, fs_q, upd_u, NUC, NQC, cewU, 1, 1);
        agg(m_uq, mx2, iv2, fs_u, upd_q, NQC, NUC, nullptr, 2, 0);  // softmax of 1 elem == 1

        // a_W transform
        gemm(upd_u, aW,               aB,        trans_u, MU, HIDC, HIDC, 0);
        gemm(upd_q, aW + HIDC * HIDC, aB + HIDC, trans_q, MQ, HIDC, HIDC, 0);

        // skip + LayerNorm
        skipln(trans_u, hu_in, sk + 0, lnG,        lnB,        hu_out, MU);
        skipln(trans_q, hq_in, sk + 1, lnG + HIDC, lnB + HIDC, hq_out, MQ);

        // ping-pong
        float* t;
        t = hu_in; hu_in = hu_out; hu_out = t;
        t = hq_in; hq_in = hq_out; hq_out = t;
    }

    // ---- output linears ----
    gemm(hu_in, in(I_OUT_W),               in(I_OUT_B),        o_utt, MU, HIDC, HIDC, 0);
    gemm(hq_in, in(I_OUT_W) + HIDC * HIDC, in(I_OUT_B) + HIDC, o_q,   MQ, HIDC, HIDC, 0);
}